// RGCN_Attn_BlockLayer_20109036879936
// MI455X (gfx1250) — compile-verified
//
#include <hip/hip_runtime.h>
#include <hip/hip_bf16.h>
#include <stdint.h>

#define N_NODES 20000
#define N_EDGES 320000
#define IN_FEAT 128
#define OUT_FEAT 128
#define NUM_RELS 64
#define NUM_BASES 4
#define NUM_HEADS 2
#define ATTN_FEAT (OUT_FEAT / NUM_HEADS)   // 64
#define SUB_IN (IN_FEAT / NUM_BASES)       // 32
#define SUB_OUT (OUT_FEAT / NUM_BASES)     // 32
#define XCOLS 384                          // [b1 head0 | b1 head1 | loop]
#define NCT (XCOLS / 16)                   // 24 col tiles of X
#define WCT (OUT_FEAT / 16)                // 8 col tiles of Bw

typedef float v2f __attribute__((ext_vector_type(2)));
typedef float v8f __attribute__((ext_vector_type(8)));
typedef int   vsi4 __attribute__((vector_size(16)));
typedef __attribute__((address_space(1))) vsi4 gvsi4;
typedef __attribute__((address_space(3))) vsi4 lvsi4;

#if defined(__HIP_DEVICE_COMPILE__) && __has_builtin(__builtin_amdgcn_global_load_async_to_lds_b128)
#define HAVE_ASYNC_LDS 1
#else
#define HAVE_ASYNC_LDS 0
#endif

// ---- monotone uint encoding of float for atomicMax-based segment max ----
__device__ __forceinline__ unsigned f2key(float f) {
    unsigned u = __float_as_uint(f);
    return (u & 0x80000000u) ? ~u : (u | 0x80000000u);
}
__device__ __forceinline__ float key2f(unsigned k) {
    unsigned u = (k & 0x80000000u) ? (k ^ 0x80000000u) : ~k;
    return __uint_as_float(u);
}

// ---------------------------------------------------------------------------
// K0: build combined matrices, pre-swizzled into WMMA B-fragment order.
//   Logical Bc[128][384]: cols [0,256) = C_h = attn_k[h] @ attn_q[h]^T
//                          cols [256,384) = loop_weight
//   Logical Bw[128][128]: Bw[o][f + 64*h] = attn_w[h][o][f]
//   Fragment order: frag[((k4*T + t)*32 + lane)] = {B[r][c], B[r+1][c]}
//     with r = k4*4 + 2*(lane>>4), c = t*16 + (lane&15).
// ---------------------------------------------------------------------------
__global__ void build_mats(const float* __restrict__ attn_k,
                           const float* __restrict__ attn_q,
                           const float* __restrict__ attn_w,
                           const float* __restrict__ loop_w,
                           float* __restrict__ Bcf, float* __restrict__ Bwf) {
    int tid = blockIdx.x * blockDim.x + threadIdx.x;
    const int nbc = 32 * NCT * 32 * 2;     // 49152
    const int nbw = 32 * WCT * 32 * 2;     // 16384
    if (tid < nbc) {
        int pair = tid & 1;
        int lane = (tid >> 1) & 31;
        int rest = tid >> 6;
        int t = rest % NCT, k4 = rest / NCT;
        int row = k4 * 4 + 2 * (lane >> 4) + pair;
        int col = t * 16 + (lane & 15);
        float v;
        if (col < 256) {
            int hh = col >> 7, o = col & 127;
            const float* K = attn_k + (hh * IN_FEAT + row) * OUT_FEAT;
            const float* Q = attn_q + (hh * OUT_FEAT + o) * OUT_FEAT;
            float s = 0.f;
            for (int p = 0; p < OUT_FEAT; ++p) s += K[p] * Q[p];
            v = s;
        } else {
            v = loop_w[row * OUT_FEAT + (col - 256)];
        }
        Bcf[tid] = v;
    } else if (tid < nbc + nbw) {
        int u = tid - nbc;
        int pair = u & 1;
        int lane = (u >> 1) & 31;
        int rest = u >> 6;
        int t = rest % WCT, k4 = rest / WCT;
        int row = k4 * 4 + 2 * (lane >> 4) + pair;
        int col = t * 16 + (lane & 15);
        int hh = col >> 6, f = col & 63;
        Bwf[u] = attn_w[(hh * OUT_FEAT + row) * ATTN_FEAT + f];
    }
}

// K0b: init segment-reduction buffers + zero the message accumulator
__global__ void init_seg(unsigned* __restrict__ amaxkey, float* __restrict__ z,
                         float* __restrict__ node) {
    long t = (long)blockIdx.x * blockDim.x + threadIdx.x;
    if (t < N_NODES * NUM_HEADS) { amaxkey[t] = 0u; z[t] = 0.f; }
    if (t < (long)N_NODES * OUT_FEAT) node[t] = 0.f;
}

// ---------------------------------------------------------------------------
// K1: node GEMM  X[N,384] = h[N,128] @ Bc via V_WMMA_F32_16X16X4_F32.
// block = 256 (8 waves); grid = (N/16, 3); wave w -> col tile blockIdx.y*8+w.
// Single-stride epilogue (no divergent branches).
// ---------------------------------------------------------------------------
__global__ void gemm_node(const float* __restrict__ h, const float* __restrict__ Bcf,
                          float* __restrict__ X) {
    __shared__ float hs[16][IN_FEAT];
    const int r0 = blockIdx.x * 16;
    for (int t = threadIdx.x; t < 16 * IN_FEAT; t += blockDim.x)
        hs[t >> 7][t & 127] = h[(long)(r0 + (t >> 7)) * IN_FEAT + (t & 127)];
    __syncthreads();

    const int wave = threadIdx.x >> 5;
    const int lane = threadIdx.x & 31;
    const int ct = blockIdx.y * 8 + wave;      // 0..23
    const int m  = lane & 15;
    const int kh = lane >> 4;
    const int n  = lane & 15;
    const v2f* Bv = (const v2f*)Bcf;

    v8f acc = {};
    for (int kk = 0; kk < IN_FEAT; kk += 4) {
        v2f a, b;
        a.x = hs[m][kk + 2 * kh + 0];
        a.y = hs[m][kk + 2 * kh + 1];
        b = Bv[((kk >> 2) * NCT + ct) * 32 + lane];
        acc = __builtin_amdgcn_wmma_f32_16x16x4_f32(false, a, false, b,
                                                    (short)0, acc, false, false);
    }
    const int gcol = ct * 16 + n;
    for (int v = 0; v < 8; ++v)
        X[(long)(r0 + v + 8 * kh) * XCOLS + gcol] = acc[v];
}

// ---------------------------------------------------------------------------
// K2: edge pass 1. One wave handles 16 edges.
//   ev[e,:] = blockdiag(W[etype[e]]) applied to h[src[e]]   (VALU, L2-hot W)
//   store ev tile; logits a[e,h] = ev . X[dst[e], h*128 : h*128+128]
//   atomicMax segment max over dst (ordered-uint keys)
// block = 128 (4 waves), grid = E/64.
// ---------------------------------------------------------------------------
__global__ void edge_pass1(const float* __restrict__ h, const float* __restrict__ W,
                           const float* __restrict__ X,
                           const int* __restrict__ src, const int* __restrict__ dst,
                           const int* __restrict__ etype,
                           float* __restrict__ evw, float* __restrict__ alog,
                           unsigned* __restrict__ amaxkey) {
    __shared__ float ev[4][16][OUT_FEAT];
    const int wave = threadIdx.x >> 5;
    const int lane = threadIdx.x & 31;
    const long e0 = (long)(blockIdx.x * 4 + wave) * 16;

    const int bblk = lane >> 3;          // which 32x32 diagonal block
    const int oi   = (lane & 7) * 4;     // 4 consecutive outputs inside block

    for (int m = 0; m < 16; ++m) {
        const int  s  = src[e0 + m];
        const int  et = etype[e0 + m];
        const float* hrow = h + (long)s * IN_FEAT + bblk * SUB_IN;
        const float* Wm   = W + (long)et * (NUM_BASES * SUB_IN * SUB_OUT)
                              + bblk * (SUB_IN * SUB_OUT);
        float a0 = 0.f, a1 = 0.f, a2 = 0.f, a3 = 0.f;
        for (int i = 0; i < SUB_IN; ++i) {
            float hv = hrow[i];                       // broadcast within b-group
            const float4 wv = *(const float4*)(Wm + i * SUB_OUT + oi);
            a0 += hv * wv.x; a1 += hv * wv.y; a2 += hv * wv.z; a3 += hv * wv.w;
        }
        const int ob = bblk * SUB_OUT + oi;
        ev[wave][m][ob + 0] = a0; ev[wave][m][ob + 1] = a1;
        ev[wave][m][ob + 2] = a2; ev[wave][m][ob + 3] = a3;
    }
    __syncthreads();

    // coalesced spill of ev tile to workspace
    for (int t = lane; t < 16 * (OUT_FEAT / 4); t += 32) {
        int m = t >> 5, q = t & 31;
        ((float4*)(evw + (e0 + m) * OUT_FEAT))[q] = ((float4*)&ev[wave][m][0])[q];
    }

    // attention logits + segment max
    for (int m = 0; m < 16; ++m) {
        const int d = dst[e0 + m];
        const float* bp = X + (long)d * XCOLS;
        for (int hh = 0; hh < NUM_HEADS; ++hh) {
            float p = 0.f;
            for (int j = 0; j < 4; ++j) {
                int o = lane + 32 * j;
                p += ev[wave][m][o] * bp[hh * 128 + o];
            }
            for (int off = 16; off; off >>= 1) p += __shfl_xor(p, off, 32);
            if (lane == 0) {
                alog[(e0 + m) * 2 + hh] = p;
                atomicMax(&amaxkey[d * 2 + hh], f2key(p));
            }
        }
    }
}

// K3: scores = exp(a - amax[dst]); z[dst] += scores (in-place over alog)
__global__ void edge_softmax_partial(const int* __restrict__ dst,
                                     const unsigned* __restrict__ amaxkey,
                                     float* __restrict__ alog, float* __restrict__ z) {
    long t = (long)blockIdx.x * blockDim.x + threadIdx.x;
    if (t >= (long)N_EDGES * 2) return;
    long e = t >> 1; int hh = (int)(t & 1);
    int d = dst[e];
    float s = expf(alog[t] - key2f(amaxkey[d * 2 + hh]));
    alog[t] = s;
    atomicAdd(&z[d * 2 + hh], s);
}

// ---------------------------------------------------------------------------
// K4: edge pass 2. te[16,128] = ev_tile @ Bw via WMMA (8 accumulators),
//     scale by score/z, atomic scatter into node[dst].
//     ev tile staged to LDS via GLOBAL_LOAD_ASYNC_TO_LDS_B128 when available.
// block = 128 (4 waves), grid = E/64.
// ---------------------------------------------------------------------------
__global__ void edge_pass2(const float* __restrict__ evw, const float* __restrict__ Bwf,
                           const float* __restrict__ scores, const float* __restrict__ z,
                           const int* __restrict__ dst, float* __restrict__ node) {
    __shared__ float ev[4][16][OUT_FEAT];
    __shared__ float coef[4][16][NUM_HEADS];
    __shared__ int   dsts[4][16];
    const int wave = threadIdx.x >> 5;
    const int lane = threadIdx.x & 31;
    const long e0 = (long)(blockIdx.x * 4 + wave) * 16;

#if HAVE_ASYNC_LDS
    for (int t = lane; t < 16 * (OUT_FEAT / 4); t += 32) {
        int m = t >> 5, q = t & 31;
        __builtin_amdgcn_global_load_async_to_lds_b128(
            (gvsi4*)(evw + (e0 + m) * OUT_FEAT + q * 4),
            (lvsi4*)&ev[wave][m][q * 4],
            0, 0);
    }
#else
    for (int t = lane; t < 16 * (OUT_FEAT / 4); t += 32) {
        int m = t >> 5, q = t & 31;
        ((float4*)&ev[wave][m][0])[q] = ((const float4*)(evw + (e0 + m) * OUT_FEAT))[q];
    }
#endif
    if (lane < 16) {
        int d = dst[e0 + lane];
        dsts[wave][lane] = d;
        coef[wave][lane][0] = scores[(e0 + lane) * 2 + 0] / z[d * 2 + 0];
        coef[wave][lane][1] = scores[(e0 + lane) * 2 + 1] / z[d * 2 + 1];
    }
#if HAVE_ASYNC_LDS
#if __has_builtin(__builtin_amdgcn_s_wait_asynccnt)
    __builtin_amdgcn_s_wait_asynccnt(0);
#else
    asm volatile("s_wait_asynccnt 0x0" ::: "memory");
#endif
#endif
    __syncthreads();

    const int m16 = lane & 15, kh = lane >> 4, n = lane & 15;
    const v2f* Bv = (const v2f*)Bwf;
    v8f acc[8] = {};
    for (int kk = 0; kk < OUT_FEAT; kk += 4) {
        v2f a;
        a.x = ev[wave][m16][kk + 2 * kh + 0];
        a.y = ev[wave][m16][kk + 2 * kh + 1];
        for (int t2 = 0; t2 < 8; ++t2) {
            v2f b = Bv[((kk >> 2) * WCT + t2) * 32 + lane];
            acc[t2] = __builtin_amdgcn_wmma_f32_16x16x4_f32(false, a, false, b,
                                                            (short)0, acc[t2], false, false);
        }
    }
    for (int t2 = 0; t2 < 8; ++t2) {
        const int col = t2 * 16 + n;
        const int hh  = col >> 6;
        for (int v = 0; v < 8; ++v) {
            int m = v + 8 * kh;                       // lanes 0-15: row v; 16-31: v+8
            float val = acc[t2][v] * coef[wave][m][hh];
            atomicAdd(&node[(long)dsts[wave][m] * OUT_FEAT + col], val);
        }
    }
}

// K5: out = relu(node + loop_message)   (loop msg lives in X cols [256,384))
__global__ void relu_out(const float* __restrict__ node, const float* __restrict__ X,
                         float* __restrict__ out) {
    long t = (long)blockIdx.x * blockDim.x + threadIdx.x;
    if (t < (long)N_NODES * OUT_FEAT) {
        long row = t >> 7; int c = (int)(t & 127);
        out[t] = fmaxf(node[t] + X[row * XCOLS + 256 + c], 0.f);
    }
}

// ---------------------------------------------------------------------------
extern "C" void kernel_launch(void* const* d_in, const int* in_sizes, int n_in,
                              void* d_out, int out_size, void* d_ws, size_t ws_size,
                              hipStream_t stream) {
    (void)in_sizes; (void)n_in; (void)out_size; (void)ws_size;
    const float* h       = (const float*)d_in[0];
    const float* weight  = (const float*)d_in[1];
    const float* attn_k  = (const float*)d_in[2];
    const float* attn_q  = (const float*)d_in[3];
    const float* attn_w  = (const float*)d_in[4];
    const float* loop_w  = (const float*)d_in[5];
    const int*   src     = (const int*)d_in[6];
    const int*   dst     = (const int*)d_in[7];
    const int*   etype   = (const int*)d_in[8];
    float* out = (float*)d_out;

    // workspace carve-up (floats)
    float* ws   = (float*)d_ws;
    float* Bcf  = ws;                                  // 49152
    float* Bwf  = Bcf + 32 * NCT * 32 * 2;             // 16384
    float* X    = Bwf + 32 * WCT * 32 * 2;             // N*384
    float* node = X + (long)N_NODES * XCOLS;           // N*128
    float* evw  = node + (long)N_NODES * 128;          // E*128
    float* alog = evw + (long)N_EDGES * 128;           // E*2 (logits -> scores)
    unsigned* amaxkey = (unsigned*)(alog + (long)N_EDGES * 2);  // N*2
    float* z    = (float*)(amaxkey + (long)N_NODES * 2);        // N*2

    const int total_bm = 32 * NCT * 32 * 2 + 32 * WCT * 32 * 2;
    build_mats<<<(total_bm + 255) / 256, 256, 0, stream>>>(attn_k, attn_q, attn_w,
                                                           loop_w, Bcf, Bwf);
    init_seg<<<(N_NODES * OUT_FEAT + 255) / 256, 256, 0, stream>>>(amaxkey, z, node);
    gemm_node<<<dim3(N_NODES / 16, 3), 256, 0, stream>>>(h, Bcf, X);
    edge_pass1<<<N_EDGES / 64, 128, 0, stream>>>(h, weight, X, src, dst, etype,
                                                 evw, alog, amaxkey);
    edge_softmax_partial<<<(N_EDGES * 2 + 255) / 256, 256, 0, stream>>>(dst, amaxkey,
                                                                        alog, z);
    edge_pass2<<<N_EDGES / 64, 128, 0, stream>>>(evw, Bwf, alog, z, dst, node);
    relu_out<<<(N_NODES * OUT_FEAT + 255) / 256, 256, 0, stream>>>(node, X, out);
}